// SSIMLoss_68547678044717
// MI455X (gfx1250) — compile-verified
//
#include <hip/hip_runtime.h>

// ---------------------------------------------------------------------------
// Fused SSIM loss for MI455X (gfx1250, wave32, WMMA).
// x,y: [32,1,512,512] f32.  out: scalar f32 = 1 - mean(ssim_map).
// Separable 11-tap Gaussian conv done with V_WMMA_F32_16X16X4_F32 as a
// banded-Toeplitz matmul; everything fused in one pass over HBM (~67MB).
// ---------------------------------------------------------------------------

typedef float v2f __attribute__((ext_vector_type(2)));
typedef float v8f __attribute__((ext_vector_type(8)));

#define IMG      512
#define TILE_H   32
#define TILE_W   64
#define IN_ROWS  48          // 42 used (32 + 10 halo), padded to 48 for 3 row-blocks
#define IN_COLS  76          // 74 used (64 + 10 halo) + 2 (K padding 26->28)
#define H_ROWS   48
#define H_STRIDE 76          // row stride of intermediates (bank-conflict free)
#define NQ       5           // x, y, x*x, y*y, x*y
#define C1F      1.0e-4f
#define C2F      9.0e-4f

// Gaussian taps, sigma=1.5, size=11, normalized (matches reference to ~1e-8)
__device__ __forceinline__ float band_w(int d) {
  float r = 0.0f;
  r = (d == 0)  ? 0.00102838f : r;
  r = (d == 1)  ? 0.00759876f : r;
  r = (d == 2)  ? 0.03600078f : r;
  r = (d == 3)  ? 0.10936069f : r;
  r = (d == 4)  ? 0.21300553f : r;
  r = (d == 5)  ? 0.26601172f : r;
  r = (d == 6)  ? 0.21300553f : r;
  r = (d == 7)  ? 0.10936069f : r;
  r = (d == 8)  ? 0.03600078f : r;
  r = (d == 9)  ? 0.00759876f : r;
  r = (d == 10) ? 0.00102838f : r;
  return r;
}

__device__ __forceinline__ v8f wmma4(v2f a, v2f b, v8f c) {
#if __has_builtin(__builtin_amdgcn_wmma_f32_16x16x4_f32)
  // (neg_a, A, neg_b, B, c_mod, C, reuse_a, reuse_b)
  return __builtin_amdgcn_wmma_f32_16x16x4_f32(false, a, false, b, (short)0, c,
                                               false, false);
#else
  // host-pass / fallback placeholder only
  c[0] += a[0] * b[0] + a[1] * b[1];
  return c;
#endif
}

__global__ __launch_bounds__(256) void ssim_main(const float* __restrict__ x,
                                                 const float* __restrict__ y,
                                                 float* __restrict__ partial,
                                                 int atomicMode) {
  __shared__ float sX[IN_ROWS * IN_COLS];
  __shared__ float sY[IN_ROWS * IN_COLS];
  __shared__ float sH[NQ][H_ROWS * H_STRIDE];
  __shared__ float sRed[256];

  const int tid     = threadIdx.x;
  const int lane    = tid & 31;
  const int wave    = tid >> 5;
  const int halfSel = lane >> 4;   // 0: lanes 0-15, 1: lanes 16-31
  const int nm      = lane & 15;   // N (B operand) or M (A operand) index

  // block -> (image, tileRow, tileCol)
  const int tilesX = IMG / TILE_W;                 // 8
  const int tilesY = IMG / TILE_H;                 // 16
  int t   = blockIdx.x % (tilesX * tilesY);
  int img = blockIdx.x / (tilesX * tilesY);
  int ty  = t / tilesX;
  int tx  = t % tilesX;
  int row0 = ty * TILE_H - 5;
  int col0 = tx * TILE_W - 5;
  const float* xb = x + (size_t)img * IMG * IMG;
  const float* yb = y + (size_t)img * IMG * IMG;

  // ---- stage 1: load input tiles with SAME zero padding --------------------
  for (int i = tid; i < IN_ROWS * IN_COLS; i += 256) {
    int r = i / IN_COLS, c = i % IN_COLS;
    int gr = row0 + r, gc = col0 + c;
    float xv = 0.0f, yv = 0.0f;
    if (gr >= 0 && gr < IMG && gc >= 0 && gc < IMG && r < 42 && c < 74) {
      size_t o = (size_t)gr * IMG + gc;
      xv = xb[o];
      yv = yb[o];
    }
    sX[i] = xv;
    sY[i] = yv;
  }

  // Banded Toeplitz operand, 7 K-chunks of 4. Same registers serve as the
  // horizontal-pass B matrix and the vertical-pass A matrix (transpose-
  // symmetric under the documented A/B lane layouts).
  v2f band[7];
#pragma unroll
  for (int k = 0; k < 7; ++k) {
    int j0 = 4 * k + 2 * halfSel;
    band[k].x = band_w(j0 - nm);
    band[k].y = band_w(j0 + 1 - nm);
  }

  __syncthreads();

  // ---- stage 2: horizontal conv via WMMA (A=data 16x26, B=band 26x16) ------
  for (int tI = wave; tI < 12; tI += 8) {      // 3 row-blocks x 4 col-blocks
    int rb = tI >> 2;
    int cb = tI & 3;
    int r     = rb * 16 + nm;                  // A-matrix row for this lane
    int cbase = cb * 16;
    v8f h0 = {}, h1 = {}, h2 = {}, h3 = {}, h4 = {};
#pragma unroll
    for (int k = 0; k < 7; ++k) {
      int j0   = 4 * k + 2 * halfSel;
      int base = r * IN_COLS + cbase + j0;
      float x0 = sX[base], x1 = sX[base + 1];
      float y0 = sY[base], y1 = sY[base + 1];
      v2f ax  = {x0, x1};
      v2f ay  = {y0, y1};
      v2f axx = {x0 * x0, x1 * x1};
      v2f ayy = {y0 * y0, y1 * y1};
      v2f axy = {x0 * y0, x1 * y1};
      h0 = wmma4(ax,  band[k], h0);
      h1 = wmma4(ay,  band[k], h1);
      h2 = wmma4(axx, band[k], h2);
      h3 = wmma4(ayy, band[k], h3);
      h4 = wmma4(axy, band[k], h4);
    }
    // C/D layout: VGPR i -> row i (+8 for lanes>=16), col = lane&15
#pragma unroll
    for (int i = 0; i < 8; ++i) {
      int rr = rb * 16 + i + 8 * halfSel;
      int cc = cbase + nm;
      int o  = rr * H_STRIDE + cc;
      sH[0][o] = h0[i];
      sH[1][o] = h1[i];
      sH[2][o] = h2[i];
      sH[3][o] = h3[i];
      sH[4][o] = h4[i];
    }
  }
  __syncthreads();

  // ---- stage 3: vertical conv via WMMA (A=band 16x26, B=data 26x16) --------
  float lsum = 0.0f;
  {
    int rb = wave >> 2;                        // 2 row-blocks
    int cb = wave & 3;                         // 4 col-blocks
    int cc = cb * 16 + nm;                     // B-matrix column for this lane
    v8f a0 = {}, a1 = {}, a2 = {}, a3 = {}, a4 = {};
#pragma unroll
    for (int k = 0; k < 7; ++k) {
      int j0   = 4 * k + 2 * halfSel;
      int base = (rb * 16 + j0) * H_STRIDE + cc;
      v2f b0 = {sH[0][base], sH[0][base + H_STRIDE]};
      v2f b1 = {sH[1][base], sH[1][base + H_STRIDE]};
      v2f b2 = {sH[2][base], sH[2][base + H_STRIDE]};
      v2f b3 = {sH[3][base], sH[3][base + H_STRIDE]};
      v2f b4 = {sH[4][base], sH[4][base + H_STRIDE]};
      a0 = wmma4(band[k], b0, a0);
      a1 = wmma4(band[k], b1, a1);
      a2 = wmma4(band[k], b2, a2);
      a3 = wmma4(band[k], b3, a3);
      a4 = wmma4(band[k], b4, a4);
    }
    // SSIM map, elementwise on the 8 accumulator slots (layout-agnostic)
#pragma unroll
    for (int i = 0; i < 8; ++i) {
      float mx = a0[i], my = a1[i];
      float exx = a2[i], eyy = a3[i], exy = a4[i];
      float sx  = exx - mx * mx;
      float sy  = eyy - my * my;
      float sxy = exy - mx * my;
      float num = (2.0f * mx * my + C1F) * (2.0f * sxy + C2F);
      float den = (mx * mx + my * my + C1F) * (sx + sy + C2F);
      float s   = num / (den + 1e-12f);
      s = fminf(fmaxf(s, -1.0f + 1e-6f), 1.0f - 1e-6f);
      lsum += s;
    }
  }
  sRed[tid] = lsum;
  __syncthreads();
  for (int off = 128; off > 0; off >>= 1) {
    if (tid < off) sRed[tid] += sRed[tid + off];
    __syncthreads();
  }
  if (tid == 0) {
    if (atomicMode) atomicAdd(partial, sRed[0]);
    else            partial[blockIdx.x] = sRed[0];
  }
}

__global__ void ssim_zero(float* p) { p[0] = 0.0f; }

__global__ __launch_bounds__(256) void ssim_final(const float* __restrict__ partial,
                                                  int n, float invN,
                                                  float* __restrict__ out) {
  __shared__ float red[256];
  float s = 0.0f;
  for (int i = threadIdx.x; i < n; i += 256) s += partial[i];
  red[threadIdx.x] = s;
  __syncthreads();
  for (int off = 128; off > 0; off >>= 1) {
    if (threadIdx.x < off) red[threadIdx.x] += red[threadIdx.x + off];
    __syncthreads();
  }
  if (threadIdx.x == 0) out[0] = 1.0f - red[0] * invN;
}

extern "C" void kernel_launch(void* const* d_in, const int* in_sizes, int n_in,
                              void* d_out, int out_size, void* d_ws, size_t ws_size,
                              hipStream_t stream) {
  (void)n_in; (void)out_size;
  const float* x = (const float*)d_in[0];
  const float* y = (const float*)d_in[1];
  float* out = (float*)d_out;
  float* ws  = (float*)d_ws;

  int nimg    = in_sizes[0] / (IMG * IMG);                   // 32
  int tiles   = (IMG / TILE_H) * (IMG / TILE_W);             // 128
  int nblocks = nimg * tiles;                                // 4096
  float invN  = 1.0f / ((float)nimg * (float)(IMG * IMG));

  if (ws_size >= (size_t)nblocks * sizeof(float)) {
    // Deterministic path: fixed per-block partials + fixed-order reduction.
    ssim_main<<<nblocks, 256, 0, stream>>>(x, y, ws, 0);
    ssim_final<<<1, 256, 0, stream>>>(ws, nblocks, invN, out);
  } else {
    // Fallback if scratch is tiny.
    ssim_zero<<<1, 1, 0, stream>>>(ws);
    ssim_main<<<nblocks, 256, 0, stream>>>(x, y, ws, 1);
    ssim_final<<<1, 256, 0, stream>>>(ws, 1, invN, out);
  }
}